// GATLayerAndWeight_1932735283943
// MI455X (gfx1250) — compile-verified
//
#include <hip/hip_runtime.h>
#include <math.h>

// ---------------------------------------------------------------------------
// GAT layer forward, MI455X (gfx1250), wave32 + V_WMMA_F32_16X16X4_F32.
// Shapes: B=16, N=128, IN_F=128, OUT_F=64, E_DIM=64, BI=BO=16, L1=64.
//
// Key algebraic collapse: the tile/cat/reshape "atom_trans" means
//   i<64 : e[b,i,j,:] = (P+Q)[b, 2i + (j>=64)]
//   i>=64: e[b,i,j,:] = P[b,(2j)%128] + Q[b,(2j+1)%128]
// with P = h@a[0:64,:], Q = h@a[64:128,:].  After lrelu + @fc_w[0:64] this is
// just two B*N scalar tables S_u / S_v.  Only the bond branch is O(B*N^2).
// ---------------------------------------------------------------------------

typedef float v2f __attribute__((ext_vector_type(2)));
typedef float v8f __attribute__((ext_vector_type(8)));

#define WMMA_F32X4(av, bv, cv) \
  __builtin_amdgcn_wmma_f32_16x16x4_f32(false, (av), false, (bv), (short)0, (cv), false, false)

__device__ __forceinline__ float lrelu01(float x) { return x > 0.0f ? x : 0.01f * x; }

// --------------------------------------------------------------------------
// Kernel 1: h = x @ W        ([16*128,128] @ [128,64])
// One wave per 16x16 output tile. grid = 16 * 8 * 4 = 512 waves.
// --------------------------------------------------------------------------
__global__ __launch_bounds__(32) void k_h(const float* __restrict__ x,
                                          const float* __restrict__ W,
                                          float* __restrict__ h) {
  const int wid   = blockIdx.x;          // 0..511
  const int b     = wid >> 5;            // /32
  const int itile = (wid >> 2) & 7;
  const int ftile = wid & 3;
  const int lane  = threadIdx.x;
  const int mr = lane & 15;              // M (for A) == N (for B) sub-index
  const int kh = lane >> 4;              // K half selector

  const float* xrow = x + ((b * 128 + itile * 16 + mr) * 128);
  const float* Wc   = W + ftile * 16 + mr;

  v8f acc = {};
  for (int k0 = 0; k0 < 128; k0 += 4) {
    const int k = k0 + 2 * kh;
    v2f av; av.x = xrow[k];          av.y = xrow[k + 1];
    v2f bv; bv.x = Wc[k * 64];       bv.y = Wc[(k + 1) * 64];
    acc = WMMA_F32X4(av, bv, acc);
  }
  float* hb = h + (b * 128 + itile * 16) * 64 + ftile * 16;
  const int n  = lane & 15;
  const int mb = (lane >> 4) * 8;
#pragma unroll
  for (int v = 0; v < 8; ++v) hb[(mb + v) * 64 + n] = acc[v];
}

// --------------------------------------------------------------------------
// Kernel 2: P = h @ a[0:64,:],  Q = h @ a[64:128,:]   (each [2048,64])
// grid = 16 * 8 * 4 * 2 = 1024 waves.
// --------------------------------------------------------------------------
__global__ __launch_bounds__(32) void k_pq(const float* __restrict__ h,
                                           const float* __restrict__ a,
                                           float* __restrict__ P,
                                           float* __restrict__ Q) {
  const int wid   = blockIdx.x;          // 0..1023
  const int b     = wid >> 6;
  const int mtile = (wid >> 3) & 7;
  const int ktile = (wid >> 1) & 3;
  const int which = wid & 1;             // 0 -> P, 1 -> Q
  const int lane  = threadIdx.x;
  const int mr = lane & 15;
  const int kh = lane >> 4;

  const float* hrow = h + ((b * 128 + mtile * 16 + mr) * 64);
  const float* ac   = a + (which * 64) * 64 + ktile * 16 + mr;

  v8f acc = {};
  for (int k0 = 0; k0 < 64; k0 += 4) {
    const int k = k0 + 2 * kh;
    v2f av; av.x = hrow[k];      av.y = hrow[k + 1];
    v2f bv; bv.x = ac[k * 64];   bv.y = ac[(k + 1) * 64];
    acc = WMMA_F32X4(av, bv, acc);
  }
  float* dst = (which ? Q : P) + (b * 128 + mtile * 16) * 64 + ktile * 16;
  const int n  = lane & 15;
  const int mb = (lane >> 4) * 8;
#pragma unroll
  for (int v = 0; v < 8; ++v) dst[(mb + v) * 64 + n] = acc[v];
}

// --------------------------------------------------------------------------
// Kernel 3: scalar tables
//   S_u[b,m] = sum_k lrelu(P[b,m,k]+Q[b,m,k]) * fc_w[k]
//   S_v[b,j] = sum_k lrelu(P[b,(2j)%128,k]+Q[b,(2j+1)%128,k]) * fc_w[k]
// 2048 threads total.
// --------------------------------------------------------------------------
__global__ __launch_bounds__(256) void k_scal(const float* __restrict__ P,
                                              const float* __restrict__ Q,
                                              const float* __restrict__ fc_w,
                                              float* __restrict__ Su,
                                              float* __restrict__ Sv) {
  const int t = blockIdx.x * 256 + threadIdx.x;   // 0..2047
  const int b = t >> 7;
  const int m = t & 127;
  const float* Pb = P + b * 128 * 64;
  const float* Qb = Q + b * 128 * 64;
  const int r0 = (2 * m) & 127;
  const int r1 = (2 * m + 1) & 127;
  float su = 0.0f, sv = 0.0f;
#pragma unroll 8
  for (int k = 0; k < 64; ++k) {
    const float fw = fc_w[k];
    su += lrelu01(Pb[m * 64 + k] + Qb[m * 64 + k]) * fw;
    sv += lrelu01(Pb[r0 * 64 + k] + Qb[r1 * 64 + k]) * fw;
  }
  Su[t] = su;
  Sv[t] = sv;
}

// --------------------------------------------------------------------------
// Kernel 4: fused logits -> softmax -> att@h -> @fc_out_w + bias -> elu.
// One block per (b, 16-row tile): grid = 16*8 = 128 blocks, 256 threads.
// --------------------------------------------------------------------------
__global__ __launch_bounds__(256) void k_att(const float* __restrict__ weight,
                                             const int*   __restrict__ adj,
                                             const float* __restrict__ h,
                                             const float* __restrict__ Su,
                                             const float* __restrict__ Sv,
                                             const float* __restrict__ wfc_w,
                                             const float* __restrict__ wfc_b,
                                             const float* __restrict__ fc_w,
                                             const float* __restrict__ fc_b,
                                             const float* __restrict__ fco_w,
                                             const float* __restrict__ fco_b,
                                             float* __restrict__ out) {
  __shared__ float s_wfc[16][16];     // weight_fc_w [c][k]
  __shared__ float s_wfcb[16];
  __shared__ float s_fcw2[16];        // fc_w[64..79]
  __shared__ float s_att[16][132];    // logits -> softmax probs (padded)
  __shared__ float s_h1[16][68];      // att @ h tile (padded)

  const int b     = blockIdx.x >> 3;
  const int itile = blockIdx.x & 7;
  const int t     = threadIdx.x;
  const int wave  = t >> 5;
  const int lane  = t & 31;

  if (t < 256) s_wfc[t >> 4][t & 15] = wfc_w[t];
  if (t < 16)  { s_wfcb[t] = wfc_b[t]; s_fcw2[t] = fc_w[64 + t]; }
  const float fcb = fc_b[0];
  __syncthreads();

  // ---- attention logits for 16 rows x 128 cols -------------------------
  for (int s = 0; s < 8; ++s) {
    const int p  = t + 256 * s;       // 0..2047
    const int il = p >> 7;
    const int j  = p & 127;
    const int ig = itile * 16 + il;

    const float* wrow = weight + (((size_t)(b * 128 + ig) * 128 + j) << 4);
    float wr[16];
#pragma unroll
    for (int c = 0; c < 16; ++c) wr[c] = wrow[c];

    float wsum = 0.0f;
#pragma unroll
    for (int k = 0; k < 16; ++k) {
      float acc = s_wfcb[k];
#pragma unroll
      for (int c = 0; c < 16; ++c) acc += wr[c] * s_wfc[c][k];
      wsum += lrelu01(acc) * s_fcw2[k];
    }
    const float ep = (ig < 64) ? Su[b * 128 + 2 * ig + (j >= 64 ? 1 : 0)]
                               : Sv[b * 128 + j];
    float logit = ep + wsum + fcb;
    if (adj[(b * 128 + ig) * 128 + j] <= 0) logit = -9.0e15f;
    s_att[il][j] = logit;
  }
  __syncthreads();

  // ---- softmax over j (each wave owns 2 rows) --------------------------
  for (int r = wave; r < 16; r += 8) {
    float v0 = s_att[r][lane];
    float v1 = s_att[r][lane + 32];
    float v2 = s_att[r][lane + 64];
    float v3 = s_att[r][lane + 96];
    float mx = fmaxf(fmaxf(v0, v1), fmaxf(v2, v3));
#pragma unroll
    for (int off = 16; off > 0; off >>= 1) mx = fmaxf(mx, __shfl_xor(mx, off, 32));
    const float e0 = __expf(v0 - mx), e1 = __expf(v1 - mx);
    const float e2 = __expf(v2 - mx), e3 = __expf(v3 - mx);
    float sm = e0 + e1 + e2 + e3;
#pragma unroll
    for (int off = 16; off > 0; off >>= 1) sm += __shfl_xor(sm, off, 32);
    const float inv = 1.0f / sm;
    s_att[r][lane]      = e0 * inv;
    s_att[r][lane + 32] = e1 * inv;
    s_att[r][lane + 64] = e2 * inv;
    s_att[r][lane + 96] = e3 * inv;
  }
  __syncthreads();

  // ---- h1 = att(16x128) @ h[b](128x64): waves 0..3, one 16x16 tile each
  const int mr = lane & 15;
  const int kh = lane >> 4;
  if (wave < 4) {
    const int colb = wave * 16;
    const float* hb = h + (size_t)b * 128 * 64 + colb + mr;
    v8f acc = {};
    for (int k0 = 0; k0 < 128; k0 += 4) {
      const int k = k0 + 2 * kh;
      v2f av; av.x = s_att[mr][k];   av.y = s_att[mr][k + 1];
      v2f bv; bv.x = hb[k * 64];     bv.y = hb[(k + 1) * 64];
      acc = WMMA_F32X4(av, bv, acc);
    }
    const int n  = lane & 15;
    const int mb = (lane >> 4) * 8;
#pragma unroll
    for (int v = 0; v < 8; ++v) s_h1[mb + v][colb + n] = acc[v];
  }
  __syncthreads();

  // ---- hn = h1(16x64) @ fc_out_w(64x64) + bias, elu --------------------
  if (wave < 4) {
    const int colb = wave * 16;
    const float* wc = fco_w + colb + mr;
    v8f acc = {};
    for (int k0 = 0; k0 < 64; k0 += 4) {
      const int k = k0 + 2 * kh;
      v2f av; av.x = s_h1[mr][k];  av.y = s_h1[mr][k + 1];
      v2f bv; bv.x = wc[k * 64];   bv.y = wc[(k + 1) * 64];
      acc = WMMA_F32X4(av, bv, acc);
    }
    const int n  = lane & 15;
    const int mb = (lane >> 4) * 8;
#pragma unroll
    for (int v = 0; v < 8; ++v) {
      float val = acc[v] + fco_b[colb + n];
      val = (val > 0.0f) ? val : (__expf(val) - 1.0f);   // elu, alpha=1
      out[(size_t)(b * 128 + itile * 16 + mb + v) * 64 + colb + n] = val;
    }
  }
}

// --------------------------------------------------------------------------
extern "C" void kernel_launch(void* const* d_in, const int* in_sizes, int n_in,
                              void* d_out, int out_size, void* d_ws, size_t ws_size,
                              hipStream_t stream) {
  const float* x      = (const float*)d_in[0];   // [16,128,128]
  const int*   adj    = (const int*)  d_in[1];   // [16,128,128]
  const float* weight = (const float*)d_in[2];   // [16,128,128,16]
  const float* W      = (const float*)d_in[3];   // [128,64]
  const float* a      = (const float*)d_in[4];   // [128,64]
  const float* wfc_w  = (const float*)d_in[5];   // [16,16]
  const float* wfc_b  = (const float*)d_in[6];   // [16]
  const float* fc_w   = (const float*)d_in[7];   // [80,1]
  const float* fc_b   = (const float*)d_in[8];   // [1]
  const float* fco_w  = (const float*)d_in[9];   // [64,64]
  const float* fco_b  = (const float*)d_in[10];  // [64]
  float* out = (float*)d_out;                    // [16,128,64]

  float* ws = (float*)d_ws;
  float* h  = ws;              // 131072
  float* P  = ws + 131072;     // 131072
  float* Q  = ws + 262144;     // 131072
  float* Su = ws + 393216;     // 2048
  float* Sv = ws + 395264;     // 2048

  k_h  <<<dim3(512),  dim3(32),  0, stream>>>(x, W, h);
  k_pq <<<dim3(1024), dim3(32),  0, stream>>>(h, a, P, Q);
  k_scal<<<dim3(8),   dim3(256), 0, stream>>>(P, Q, fc_w, Su, Sv);
  k_att<<<dim3(128),  dim3(256), 0, stream>>>(weight, adj, h, Su, Sv,
                                              wfc_w, wfc_b, fc_w, fc_b,
                                              fco_w, fco_b, out);
}